// Attn_63359357551158
// MI455X (gfx1250) — compile-verified
//
#include <hip/hip_runtime.h>

#define S_LEN 2048
#define B_LEN 32
#define H_LEN 1024

typedef float v2f __attribute__((ext_vector_type(2)));
typedef float v8f __attribute__((ext_vector_type(8)));

// ---------------------------------------------------------------------------
// Kernel 1: u[b,h] = sum_k dec[b,k] * W[k,h]   (M=32, K=1024, N=1024)
// fp32 WMMA 16x16x4. 128 wave-tiles (2 M-tiles x 64 N-tiles), 16 blocks x 8 waves.
// A 16x4 layout: lane&15 = M; v0/v1 = K{0,1} (lanes 0-15) or K{2,3} (lanes 16-31).
// B 4x16 layout: lane&15 = N; v0/v1 = K{0,1} (lanes 0-15) or K{2,3} (lanes 16-31).
// D layout: VGPR v holds M = v + 8*(lane>>4), N = lane&15.
// ---------------------------------------------------------------------------
__global__ void __launch_bounds__(256) proj_wmma_kernel(
    const float* __restrict__ dec,   // [B,H]
    const float* __restrict__ W,     // [H,H], element W[k*H + h]
    float* __restrict__ u)           // [B,H]
{
    const int lane  = threadIdx.x & 31;
    const int wave  = threadIdx.x >> 5;
    const int tile  = blockIdx.x * 8 + wave;   // 0..127
    const int mTile = (tile & 1) * 16;         // b-offset: 0 or 16
    const int nTile = (tile >> 1) * 16;        // h-offset: 0..1008

    const int mn    = lane & 15;               // M for A, N for B/D
    const int khalf = lane >> 4;               // which K pair this half-wave owns

    v8f acc = {};
    const float* arow = dec + (size_t)(mTile + mn) * H_LEN + 2 * khalf;
    const float* bcol = W + nTile + mn;

    for (int k0 = 0; k0 < H_LEN; k0 += 4) {
        v2f a, b;
        a.x = arow[k0 + 0];
        a.y = arow[k0 + 1];
        const int kb = k0 + 2 * khalf;
        b.x = bcol[(size_t)(kb + 0) * H_LEN];
        b.y = bcol[(size_t)(kb + 1) * H_LEN];
        // (neg_a, A, neg_b, B, c_mod, C, reuse_a, reuse_b)
        acc = __builtin_amdgcn_wmma_f32_16x16x4_f32(false, a, false, b,
                                                    (short)0, acc, false, false);
    }

    #pragma unroll
    for (int v = 0; v < 8; ++v) {
        const int row = mTile + v + 8 * khalf;
        u[(size_t)row * H_LEN + nTile + mn] = acc[v];
    }
}

// ---------------------------------------------------------------------------
// Kernel 2: c[b] = dot(dec[b,:], b_attn)
// ---------------------------------------------------------------------------
__global__ void __launch_bounds__(256) bias_dot_kernel(
    const float* __restrict__ dec, const float* __restrict__ bias,
    float* __restrict__ c)
{
    const int b = blockIdx.x;
    float acc = 0.f;
    for (int k = threadIdx.x; k < H_LEN; k += 256)
        acc = fmaf(dec[(size_t)b * H_LEN + k], bias[k], acc);
    for (int m = 16; m; m >>= 1) acc += __shfl_xor(acc, m, 32);
    __shared__ float red[8];
    if ((threadIdx.x & 31) == 0) red[threadIdx.x >> 5] = acc;
    __syncthreads();
    if (threadIdx.x == 0) {
        float t = 0.f;
        #pragma unroll
        for (int i = 0; i < 8; ++i) t += red[i];
        c[b] = t;
    }
}

// ---------------------------------------------------------------------------
// Kernel 3 (bandwidth-bound phase): energy[b,s] = enc[s,b,:] . u[b,:] + c[b]
// Streams the 256 MB encoder tensor once with float4 loads; u[b] cached in LDS.
// grid = (S/128, B), 256 threads = 8 waves, one wave per s-row, 16 rows/wave.
// ---------------------------------------------------------------------------
__global__ void __launch_bounds__(256) energy_kernel(
    const float* __restrict__ enc,   // [S,B,H]
    const float* __restrict__ u,     // [B,H]
    const float* __restrict__ c,     // [B]
    float* __restrict__ energy)      // [B,S]
{
    const int b = blockIdx.y;
    __shared__ float ubuf[H_LEN];    // 4 KB
    for (int i = threadIdx.x; i < H_LEN; i += 256)
        ubuf[i] = u[(size_t)b * H_LEN + i];
    __syncthreads();

    const float cb   = c[b];
    const int   wave = threadIdx.x >> 5;
    const int   lane = threadIdx.x & 31;
    const int   s0   = blockIdx.x * 128;
    const float4* u4 = (const float4*)ubuf;

    for (int s = s0 + wave; s < s0 + 128; s += 8) {
        const float4* e4 = (const float4*)(enc + (size_t)s * (B_LEN * H_LEN)
                                               + (size_t)b * H_LEN);
        // hint the next row this wave will touch into cache
        __builtin_prefetch(enc + (size_t)(s + 8) * (B_LEN * H_LEN)
                               + (size_t)b * H_LEN + 4 * lane, 0, 0);
        float acc = 0.f;
        #pragma unroll
        for (int j = 0; j < 8; ++j) {          // 8 x float4 = 32 floats/lane
            const float4 e  = e4[lane + 32 * j];
            const float4 uu = u4[lane + 32 * j];
            acc = fmaf(e.x, uu.x, acc);
            acc = fmaf(e.y, uu.y, acc);
            acc = fmaf(e.z, uu.z, acc);
            acc = fmaf(e.w, uu.w, acc);
        }
        for (int m = 16; m; m >>= 1) acc += __shfl_xor(acc, m, 32);
        if (lane == 0) energy[(size_t)b * S_LEN + s] = acc + cb;
    }
}

// ---------------------------------------------------------------------------
// Kernel 4: out[b,s] = softmax_s(energy[b,s])
// ---------------------------------------------------------------------------
__global__ void __launch_bounds__(256) softmax_kernel(
    const float* __restrict__ energy, float* __restrict__ out)
{
    const int b = blockIdx.x;
    const float* e = energy + (size_t)b * S_LEN;
    float*       o = out    + (size_t)b * S_LEN;
    __shared__ float red[8];

    float m = -3.402823466e+38f;
    for (int s = threadIdx.x; s < S_LEN; s += 256)
        m = fmaxf(m, e[s]);
    for (int k = 16; k; k >>= 1) m = fmaxf(m, __shfl_xor(m, k, 32));
    if ((threadIdx.x & 31) == 0) red[threadIdx.x >> 5] = m;
    __syncthreads();
    m = red[0];
    #pragma unroll
    for (int i = 1; i < 8; ++i) m = fmaxf(m, red[i]);
    __syncthreads();   // red reused below

    float sum = 0.f;
    for (int s = threadIdx.x; s < S_LEN; s += 256)
        sum += expf(e[s] - m);
    for (int k = 16; k; k >>= 1) sum += __shfl_xor(sum, k, 32);
    if ((threadIdx.x & 31) == 0) red[threadIdx.x >> 5] = sum;
    __syncthreads();
    sum = 0.f;
    #pragma unroll
    for (int i = 0; i < 8; ++i) sum += red[i];
    const float inv = 1.0f / sum;

    for (int s = threadIdx.x; s < S_LEN; s += 256)
        o[s] = expf(e[s] - m) * inv;
}

// ---------------------------------------------------------------------------
extern "C" void kernel_launch(void* const* d_in, const int* in_sizes, int n_in,
                              void* d_out, int out_size, void* d_ws, size_t ws_size,
                              hipStream_t stream) {
    const float* dec  = (const float*)d_in[0];  // rnn_outputs [1,B,H]
    const float* enc  = (const float*)d_in[1];  // encoder_outputs [S,B,H]
    const float* W    = (const float*)d_in[2];  // W_attn [H,H]
    const float* bias = (const float*)d_in[3];  // b_attn [H]
    float* out = (float*)d_out;                 // [B,S]

    float* u      = (float*)d_ws;               // 32*1024 floats
    float* c      = u + B_LEN * H_LEN;          // 32 floats (+pad)
    float* energy = c + 64;                     // 32*2048 floats

    proj_wmma_kernel<<<16, 256, 0, stream>>>(dec, W, u);
    bias_dot_kernel<<<B_LEN, 256, 0, stream>>>(dec, bias, c);
    energy_kernel<<<dim3(S_LEN / 128, B_LEN), 256, 0, stream>>>(enc, u, c, energy);
    softmax_kernel<<<B_LEN, 256, 0, stream>>>(energy, out);
}